// RWKV_LAYER_76519137346072
// MI455X (gfx1250) — compile-verified
//
#include <hip/hip_runtime.h>
#include <math.h>

#define LD 12
#define DD 2048
#define FF 8192

typedef __attribute__((ext_vector_type(2))) float v2f;
typedef __attribute__((ext_vector_type(8))) float v8f;

// workspace layout (float offsets)
#define WS_X     0
#define WS_R     (1*DD)
#define WS_K     (2*DD)
#define WS_V     (3*DD)
#define WS_SX    (4*DD)
#define WS_FR    (5*DD)
#define WS_FK    (6*DD)     // 8192 floats (4*DD)
#define WS_PART  (10*DD)    // 4 * 2048 split-K partials
// total = 14*DD floats = 114688 bytes

__device__ __forceinline__ float sigmoidf_(float x) { return 1.f / (1.f + expf(-x)); }

// Block-wide mean / rsqrt(var+eps) over a 2048-float LDS vector (blockDim=256).
__device__ __forceinline__ void block_stats(const float* xs, float* red, float& mean, float& rstd)
{
    float s = 0.f, q = 0.f;
    for (int i = threadIdx.x; i < DD; i += 256) { float v = xs[i]; s += v; q += v * v; }
    for (int off = 16; off > 0; off >>= 1) {
        s += __shfl_down(s, off, 32);
        q += __shfl_down(q, off, 32);
    }
    const int w = threadIdx.x >> 5;
    if ((threadIdx.x & 31) == 0) { red[w] = s; red[8 + w] = q; }
    __syncthreads();
    if (threadIdx.x == 0) {
        float S = 0.f, Q = 0.f;
        for (int i = 0; i < 8; ++i) { S += red[i]; Q += red[8 + i]; }
        float m = S / (float)DD;
        float var = Q / (float)DD - m * m;
        red[16] = m;
        red[17] = rsqrtf(var + 1e-5f);
    }
    __syncthreads();
    mean = red[16];
    rstd = red[17];
}

// One wave computes 16 output rows of W(rows x klen) @ vec via V_WMMA_F32_16X16X4_F32.
// A = 16x4 weight tile (per-lane b64 nontemporal loads), B = vec chunk broadcast into
// all 16 columns (every column of D is identical). Two accumulators break the RAW chain.
__device__ __forceinline__ void wmma_matvec(const float* __restrict__ W, size_t ld,
                                            int row_base, const float* vec, int klen,
                                            float out[8])
{
    const int lane = threadIdx.x & 31;
    const int m  = lane & 15;
    const int hi = lane >> 4;   // 0: K even pair, 1: K odd pair (A); B row select
    const float* wp = W + (size_t)(row_base + m) * ld + 2 * hi;
    const float* vp = vec + hi;
    v8f c0 = {0.f, 0.f, 0.f, 0.f, 0.f, 0.f, 0.f, 0.f};
    v8f c1 = {0.f, 0.f, 0.f, 0.f, 0.f, 0.f, 0.f, 0.f};
#pragma unroll 4
    for (int k = 0; k < klen; k += 8) {
        v2f a0 = __builtin_nontemporal_load((const v2f*)(wp));
        v2f a1 = __builtin_nontemporal_load((const v2f*)(wp + 4));
        v2f b0, b1;
        b0.x = vp[0]; b0.y = vp[2];
        b1.x = vp[4]; b1.y = vp[6];
        c0 = __builtin_amdgcn_wmma_f32_16x16x4_f32(false, a0, false, b0, (short)0, c0, false, false);
        c1 = __builtin_amdgcn_wmma_f32_16x16x4_f32(false, a1, false, b1, (short)0, c1, false, false);
        wp += 8; vp += 8;
    }
#pragma unroll
    for (int g = 0; g < 8; ++g) out[g] = c0[g] + c1[g];
}

// ---------------- K1: x build + LN1 + mix + fused r/k/v matvecs ----------------
__global__ __launch_bounds__(256) void k1_rkv(
    int l, const float* __restrict__ x_in, const float* __restrict__ state,
    const float* __restrict__ ln1w, const float* __restrict__ ln1b,
    const float* __restrict__ tmk, const float* __restrict__ tmv, const float* __restrict__ tmr,
    const float* __restrict__ kw, const float* __restrict__ vw, const float* __restrict__ rw,
    float* __restrict__ ws)
{
    __shared__ float xs[DD];
    __shared__ float vec[DD];
    __shared__ float red[32];
    const int tid = threadIdx.x;
    const int bid = blockIdx.x;

    // current-layer input x (combine previous layer's split-K partials, fixed order)
    for (int i = tid; i < DD; i += 256) {
        float xi;
        if (l == 0) {
            xi = x_in[i];
        } else {
            const float* p = ws + WS_PART;
            xi = ws[WS_SX + i] + ws[WS_FR + i] * (p[i] + p[DD + i] + p[2 * DD + i] + p[3 * DD + i]);
        }
        xs[i] = xi;
        if (bid == 0) ws[WS_X + i] = xi;   // materialize for K2
    }
    __syncthreads();

    float mean, rstd;
    block_stats(xs, red, mean, rstd);

    const int mat = bid >> 4;              // 0=r, 1=k, 2=v (16 blocks each)
    const float* w1 = ln1w + l * DD;
    const float* b1 = ln1b + l * DD;
    const float* s1 = state + (size_t)(l * 5 + 1) * DD;
    const float* tm = (mat == 0 ? tmr : mat == 1 ? tmk : tmv) + l * DD;
    for (int i = tid; i < DD; i += 256) {
        float xn = (xs[i] - mean) * rstd * w1[i] + b1[i];
        float t = tm[i];
        vec[i] = xn * t + s1[i] * (1.f - t);
    }
    __syncthreads();

    const float* W = (mat == 0 ? rw : mat == 1 ? kw : vw) + (size_t)l * DD * DD;
    const int wave = tid >> 5;
    const int row_base = (bid & 15) * 128 + wave * 16;
    float acc[8];
    wmma_matvec(W, DD, row_base, vec, DD, acc);

    const int lane = tid & 31;
    if ((lane & 15) == 0) {                // lanes 0 and 16 hold rows [rb..rb+7], [rb+8..rb+15]
        float* dst = ws + (mat == 0 ? WS_R : mat == 1 ? WS_K : WS_V);
        const int r0 = row_base + 8 * (lane >> 4);
#pragma unroll
        for (int g = 0; g < 8; ++g) {
            float v = acc[g];
            if (mat == 0) v = sigmoidf_(v);
            dst[r0 + g] = v;
        }
    }
}

// ---------------- K2: WKV elementwise + ow matvec + residual + SA states ----------------
__global__ __launch_bounds__(256) void k2_sa(
    int l, const float* __restrict__ state,
    const float* __restrict__ ln1w, const float* __restrict__ ln1b,
    const float* __restrict__ tf, const float* __restrict__ td,
    const float* __restrict__ ow,
    float* __restrict__ out_state, float* __restrict__ ws)
{
    __shared__ float xs[DD];
    __shared__ float rab[DD];
    __shared__ float red[32];
    const int tid = threadIdx.x;
    const int bid = blockIdx.x;

    const float* st = state + (size_t)l * 5 * DD;
    const float* aa = st + 2 * DD;
    const float* bb = st + 3 * DD;
    const float* pp = st + 4 * DD;
    const float* tfl = tf + l * DD;
    const float* tdl = td + l * DD;
    float* ost = out_state + (size_t)l * 5 * DD;

    for (int i = tid; i < DD; i += 256) {
        xs[i] = ws[WS_X + i];
        float k = ws[WS_K + i], v = ws[WS_V + i], r = ws[WS_R + i];
        float ppi = pp[i], aai = aa[i], bbi = bb[i];
        float wwv = tfl[i] + k;
        float pm = fmaxf(ppi, wwv);
        float e1 = expf(ppi - pm), e2 = expf(wwv - pm);
        float a = e1 * aai + e2 * v;
        float b = e1 * bbi + e2;
        rab[i] = r * a / b;
        if (bid == 0) {
            float ww2 = ppi + tdl[i];
            float p2 = fmaxf(ww2, k);
            float e1b = expf(ww2 - p2), e2b = expf(k - p2);
            ost[2 * DD + i] = e1b * aai + e2b * v;   // naa
            ost[3 * DD + i] = e1b * bbi + e2b;       // nbb
            ost[4 * DD + i] = p2;                    // p2
        }
    }
    __syncthreads();

    if (bid == 0) {   // state[1] = xn = LN1(x)
        float mean, rstd;
        block_stats(xs, red, mean, rstd);
        const float* w1 = ln1w + l * DD;
        const float* b1 = ln1b + l * DD;
        for (int i = tid; i < DD; i += 256)
            ost[DD + i] = (xs[i] - mean) * rstd * w1[i] + b1[i];
    }

    const float* W = ow + (size_t)l * DD * DD;
    const int wave = tid >> 5;
    const int row_base = bid * 128 + wave * 16;
    float acc[8];
    wmma_matvec(W, DD, row_base, rab, DD, acc);

    const int lane = tid & 31;
    if ((lane & 15) == 0) {
        const int r0 = row_base + 8 * (lane >> 4);
#pragma unroll
        for (int g = 0; g < 8; ++g)
            ws[WS_SX + r0 + g] = xs[r0 + g] + acc[g];   // sx = x + ow@(r*a/b)
    }
}

// ---------------- K3: LN2 + mix + fused frw (sigmoid) / fkw (relu^2) ----------------
__global__ __launch_bounds__(256) void k3_ffn_rk(
    int l, const float* __restrict__ state,
    const float* __restrict__ ln2w, const float* __restrict__ ln2b,
    const float* __restrict__ ftmk, const float* __restrict__ ftmr,
    const float* __restrict__ frw, const float* __restrict__ fkw,
    float* __restrict__ out_state, float* __restrict__ ws)
{
    __shared__ float xs[DD];
    __shared__ float vec[DD];
    __shared__ float red[32];
    const int tid = threadIdx.x;
    const int bid = blockIdx.x;

    for (int i = tid; i < DD; i += 256) xs[i] = ws[WS_SX + i];
    __syncthreads();

    float mean, rstd;
    block_stats(xs, red, mean, rstd);

    const int mat = (bid < 16) ? 0 : 1;     // 0: frw (2048 rows), 1: fkw (8192 rows)
    const float* w2 = ln2w + l * DD;
    const float* b2 = ln2b + l * DD;
    const float* s0 = state + (size_t)l * 5 * DD;
    const float* tm = (mat == 0 ? ftmr : ftmk) + l * DD;
    float* ost = out_state + (size_t)l * 5 * DD;
    for (int i = tid; i < DD; i += 256) {
        float x2 = (xs[i] - mean) * rstd * w2[i] + b2[i];
        float t = tm[i];
        vec[i] = x2 * t + s0[i] * (1.f - t);
        if (bid == 0) ost[i] = x2;          // state[0] = x2
    }
    __syncthreads();

    const int wave = tid >> 5;
    const int lane = tid & 31;
    float acc[8];
    if (mat == 0) {
        const float* W = frw + (size_t)l * DD * DD;
        const int row_base = bid * 128 + wave * 16;
        wmma_matvec(W, DD, row_base, vec, DD, acc);
        if ((lane & 15) == 0) {
            const int r0 = row_base + 8 * (lane >> 4);
#pragma unroll
            for (int g = 0; g < 8; ++g) ws[WS_FR + r0 + g] = sigmoidf_(acc[g]);
        }
    } else {
        const float* W = fkw + (size_t)l * FF * DD;
        const int row_base = (bid - 16) * 128 + wave * 16;
        wmma_matvec(W, DD, row_base, vec, DD, acc);
        if ((lane & 15) == 0) {
            const int r0 = row_base + 8 * (lane >> 4);
#pragma unroll
            for (int g = 0; g < 8; ++g) {
                float v = fmaxf(acc[g], 0.f);
                ws[WS_FK + r0 + g] = v * v;
            }
        }
    }
}

// ---------------- K4: value_ffn (D x F) with split-K=4 -> deterministic partials ----------------
__global__ __launch_bounds__(256) void k4_ffn_v(
    int l, const float* __restrict__ fvw, float* __restrict__ ws)
{
    __shared__ float fk[DD];
    const int tid = threadIdx.x;
    const int bid = blockIdx.x;
    const int kc = bid >> 4;       // 0..3 K-chunk
    const int rowblk = bid & 15;   // 0..15 row block

    for (int i = tid; i < DD; i += 256) fk[i] = ws[WS_FK + kc * DD + i];
    __syncthreads();

    const float* W = fvw + (size_t)l * DD * FF + (size_t)kc * DD;
    const int wave = tid >> 5;
    const int row_base = rowblk * 128 + wave * 16;
    float acc[8];
    wmma_matvec(W, FF, row_base, fk, DD, acc);

    const int lane = tid & 31;
    if ((lane & 15) == 0) {
        const int r0 = row_base + 8 * (lane >> 4);
#pragma unroll
        for (int g = 0; g < 8; ++g) ws[WS_PART + kc * DD + r0 + g] = acc[g];
    }
}

// ---------------- K5: final combine -> x_out ----------------
__global__ __launch_bounds__(256) void k5_final(float* __restrict__ out, const float* __restrict__ ws)
{
    for (int i = threadIdx.x; i < DD; i += 256) {
        const float* p = ws + WS_PART;
        out[i] = ws[WS_SX + i] + ws[WS_FR + i] * (p[i] + p[DD + i] + p[2 * DD + i] + p[3 * DD + i]);
    }
}

extern "C" void kernel_launch(void* const* d_in, const int* in_sizes, int n_in,
                              void* d_out, int out_size, void* d_ws, size_t ws_size,
                              hipStream_t stream)
{
    (void)in_sizes; (void)n_in; (void)out_size; (void)ws_size;
    const float* x      = (const float*)d_in[0];
    const float* state  = (const float*)d_in[1];
    const float* ln1w   = (const float*)d_in[2];
    const float* ln1b   = (const float*)d_in[3];
    const float* ln2w   = (const float*)d_in[4];
    const float* ln2b   = (const float*)d_in[5];
    const float* td     = (const float*)d_in[6];
    const float* tf     = (const float*)d_in[7];
    const float* tmk    = (const float*)d_in[8];
    const float* tmv    = (const float*)d_in[9];
    const float* tmr    = (const float*)d_in[10];
    const float* key    = (const float*)d_in[11];
    const float* value  = (const float*)d_in[12];
    const float* rec    = (const float*)d_in[13];
    const float* outv   = (const float*)d_in[14];
    const float* ftmk   = (const float*)d_in[15];
    const float* ftmr   = (const float*)d_in[16];
    const float* keyf   = (const float*)d_in[17];
    const float* recf   = (const float*)d_in[18];
    const float* valf   = (const float*)d_in[19];

    float* out = (float*)d_out;
    float* out_state = out + DD;          // new_states start after x_out
    float* ws = (float*)d_ws;

    for (int l = 0; l < LD; ++l) {
        k1_rkv<<<48, 256, 0, stream>>>(l, x, state, ln1w, ln1b, tmk, tmv, tmr, key, value, rec, ws);
        k2_sa<<<16, 256, 0, stream>>>(l, state, ln1w, ln1b, tf, td, outv, out_state, ws);
        k3_ffn_rk<<<80, 256, 0, stream>>>(l, state, ln2w, ln2b, ftmk, ftmr, recf, keyf, out_state, ws);
        k4_ffn_v<<<64, 256, 0, stream>>>(l, valf, ws);
    }
    k5_final<<<1, 256, 0, stream>>>(out, ws);
}